// Channel_Wise_DiffLoss_8693013807330
// MI455X (gfx1250) — compile-verified
//
#include <hip/hip_runtime.h>

// ---------------------------------------------------------------------------
// Channel-wise DiffLoss, factorized:
//   out = ( (1/2048^2) * sum_p  s1[p]*s2[p] )^2
//   s[p] = sum_c x[p,c] * r[c],   r[c] = 1 / (||col_c||_2 + 1e-6)
// Shapes: x viewed as [P=16384, C=2048] fp32 (raw reshape of (8,256,128,128)).
// Memory-bound: ~536 MB traffic -> ~23 us at 23.3 TB/s. The weighted row
// reduction runs on the WMMA (XDL) pipe: A = raw 16x4 f32 tile, B = r-chunk.
// ---------------------------------------------------------------------------

typedef __attribute__((ext_vector_type(2))) float v2f;
typedef __attribute__((ext_vector_type(8))) float v8f;
typedef __attribute__((ext_vector_type(4))) int   v4i;

// AS-qualified pointer types for the gfx1250 async global->LDS builtin:
// param0 = global (addrspace 1) v4i*, param1 = LDS (addrspace 3) v4i*.
typedef __attribute__((address_space(1))) v4i* gas_v4i_ptr;
typedef __attribute__((address_space(3))) v4i* lds_v4i_ptr;

#define COLS   2048
#define PIX    16384
#define CHUNK  64            // columns staged per LDS buffer
#define RSTRIDE 68           // 64 + 4 pad floats -> bank-conflict-free tile reads
#define WAVES  8             // 256-thread workgroup = 8 wave32

#define EPS 1e-6f

// ---- gfx1250 async global->LDS copy (guarded; fallback keeps compile clean)
__device__ __forceinline__ void async_copy_b128(const float* gp, float* lp) {
#if __has_builtin(__builtin_amdgcn_global_load_async_to_lds_b128)
    __builtin_amdgcn_global_load_async_to_lds_b128(
        (gas_v4i_ptr)(void*)const_cast<float*>(gp),
        (lds_v4i_ptr)(void*)lp,
        0, 0);
#else
    *(float4*)lp = *(const float4*)gp;   // global_load_b128 + ds_store_b128
#endif
}

__device__ __forceinline__ void wait_async0() {
#if __has_builtin(__builtin_amdgcn_s_wait_asynccnt)
    __builtin_amdgcn_s_wait_asynccnt(0);
#else
    asm volatile("s_wait_asynccnt 0" ::: "memory");
#endif
    asm volatile("" ::: "memory");
}

// ---------------------------------------------------------------------------
// Kernel 0: zero the accumulator region of the workspace.
// ws layout (floats): [0..4095] sq1|sq2, [4096..8191] r1|r2, [8192] dot acc
// ---------------------------------------------------------------------------
__global__ void k_zero(float* ws) {
    int i = blockIdx.x * 256 + threadIdx.x;
    if (i <= 8192) ws[i] = 0.0f;
}

// ---------------------------------------------------------------------------
// Kernel 1: per-column sum of squares for both inputs.
// grid = (8, 64, 2), block = 256. Thread t of block (bx,by,bz) owns column
// bx*256+t over a 256-row slab -> fully coalesced 128B wave requests.
// ---------------------------------------------------------------------------
__global__ void k_colsq(const float* __restrict__ x1,
                        const float* __restrict__ x2,
                        float* __restrict__ sq) {
    const int col  = blockIdx.x * 256 + threadIdx.x;
    const int row0 = blockIdx.y * 256;
    const float* x = blockIdx.z ? x2 : x1;
    const float* p = x + (size_t)row0 * COLS + col;
    float acc = 0.0f;
#pragma unroll 4
    for (int i = 0; i < 256; ++i) {
        float v = p[(size_t)i * COLS];
        acc += v * v;
    }
    atomicAdd(&sq[blockIdx.z * COLS + col], acc);
}

// ---------------------------------------------------------------------------
// Kernel 2: r[c] = 1 / (sqrt(sq[c]) + eps)   (4096 values)
// ---------------------------------------------------------------------------
__global__ void k_recip(const float* __restrict__ sq, float* __restrict__ r) {
    int i = blockIdx.x * 256 + threadIdx.x;
    if (i < 2 * COLS) r[i] = 1.0f / (sqrtf(sq[i]) + EPS);
}

// ---------------------------------------------------------------------------
// Kernel 3: dot += sum_p s1[p]*s2[p] via V_WMMA_F32_16X16X4_F32.
// Each wave owns a 16-row tile; stages 16x64 fp32 chunks into its private LDS
// buffer with async b128 copies, then feeds 16 WMMAs per chunk:
//   A = 16x4 raw x tile (ds_load_b64 per documented A layout),
//   B[k,n] = r[cbase+k]  (the /norm weighting runs inside the matrix op).
// D accumulates s[p] replicated across N (C/D layout: vgpr v, lane<16 -> M=v,
// lane>=16 -> M=v+8).
// ---------------------------------------------------------------------------
__global__ void __launch_bounds__(256)
k_rowdot(const float* __restrict__ x1, const float* __restrict__ x2,
         const float* __restrict__ r, float* __restrict__ acc) {
    __shared__ float sh[WAVES * 16 * RSTRIDE + 2 * COLS];   // 51,200 B

    const int tid  = threadIdx.x;
    const int wave = tid >> 5;
    const int lane = tid & 31;
    const int half = lane >> 4;     // 0: K={0,1}, 1: K={2,3}
    const int mrow = lane & 15;

    float* rtab = &sh[WAVES * 16 * RSTRIDE];
    for (int i = tid; i < 2 * COLS; i += 256) rtab[i] = r[i];
    __syncthreads();

    float* buf = &sh[wave * 16 * RSTRIDE];
    const int tile_row = (blockIdx.x * WAVES + wave) * 16;

    v8f c1 = {0.f,0.f,0.f,0.f,0.f,0.f,0.f,0.f};
    v8f c2 = {0.f,0.f,0.f,0.f,0.f,0.f,0.f,0.f};

    for (int which = 0; which < 2; ++which) {
        const float* x  = which ? x2 : x1;
        const float* rt = rtab + which * COLS;
        v8f cacc = {0.f,0.f,0.f,0.f,0.f,0.f,0.f,0.f};

        for (int c0 = 0; c0 < COLS; c0 += CHUNK) {
            // stage 16 rows x 64 cols = 256 b128 transfers, 8 per lane
#pragma unroll
            for (int j = 0; j < 8; ++j) {
                int i   = lane + 32 * j;
                int rr  = i >> 4;       // 16 segments per row
                int seg = i & 15;
                const float* gp = x + (size_t)(tile_row + rr) * COLS + c0 + seg * 4;
                float*       lp = buf + rr * RSTRIDE + seg * 4;
                async_copy_b128(gp, lp);
            }
            wait_async0();

#pragma unroll
            for (int kk = 0; kk < CHUNK; kk += 4) {
                // A: lane<16 -> row=lane, K={kk,kk+1}; lane>=16 -> K={kk+2,kk+3}
                v2f a = *(const v2f*)(buf + mrow * RSTRIDE + kk + 2 * half);
                // B: broadcast r values along N; row split mirrors A's K split
                v2f b = *(const v2f*)(rt + c0 + kk + 2 * half);
                cacc = __builtin_amdgcn_wmma_f32_16x16x4_f32(
                           false, a, false, b, (short)0, cacc, false, false);
            }
        }
        if (which == 0) c1 = cacc; else c2 = cacc;
    }

    // per-lane partial: rows 0-7 (lanes 0-15) or rows 8-15 (lanes 16-31)
    float p = 0.0f;
#pragma unroll
    for (int v = 0; v < 8; ++v) p += c1[v] * c2[v];
    p += __shfl_xor(p, 16, 32);          // combine the two row halves
    if (lane == 0) atomicAdd(acc, p);
}

// ---------------------------------------------------------------------------
// Kernel 4: out = ( acc / 2048^2 )^2
// ---------------------------------------------------------------------------
__global__ void k_final(const float* __restrict__ acc, float* __restrict__ out) {
    float m = acc[0] * (1.0f / ((float)COLS * (float)COLS));
    out[0] = m * m;
}

extern "C" void kernel_launch(void* const* d_in, const int* in_sizes, int n_in,
                              void* d_out, int out_size, void* d_ws, size_t ws_size,
                              hipStream_t stream) {
    const float* x1 = (const float*)d_in[0];
    const float* x2 = (const float*)d_in[1];
    float* out = (float*)d_out;
    float* ws  = (float*)d_ws;

    float* sq  = ws;           // 4096 floats
    float* r   = ws + 4096;    // 4096 floats
    float* acc = ws + 8192;    // 1 float

    k_zero  <<<(8193 + 255) / 256, 256, 0, stream>>>(ws);
    k_colsq <<<dim3(8, 64, 2),     256, 0, stream>>>(x1, x2, sq);
    k_recip <<<(2 * COLS + 255) / 256, 256, 0, stream>>>(sq, r);
    k_rowdot<<<PIX / (16 * WAVES), 256, 0, stream>>>(x1, x2, r, acc);
    k_final <<<1, 1, 0, stream>>>(acc, out);
}